// HashTable_86105504350248
// MI455X (gfx1250) — compile-verified
//
#include <hip/hip_runtime.h>
#include <math.h>
#include <stdint.h>
#include <stddef.h>

// ---------------------------------------------------------------------------
// Multi-LOD 2D hash-grid encode (Instant-NGP style) for MI455X / gfx1250.
//
// Reference: for each of 2^21 points and 16 LODs, bilinear-interp 4 gathered
// float2 features from a (16, 2^19, 2) f32 codebook; output (2^21, 32) f32.
//
// Memory-bound design (no matrix math exists in this op):
//  * thread == (point, lod): out address is exactly out + 2*tid -> perfectly
//    coalesced float2 stores, no LDS transpose needed.
//  * nontemporal stores for the 256MB write-once output stream (TH_STORE_NT)
//    so the 64MB codebook stays resident in the 192MB L2.
//  * global_prefetch_b8 warm pass folded into the first 2048 blocks: streams
//    the whole 64MB codebook into L2 ahead of the random 8B gathers.
//  * hash levels: power-of-two grid -> (x*P0 ^ y*P1) & 0x7FFFF (floored mod
//    by 2^19 == bitwise and, matching JAX int32 semantics).
// ---------------------------------------------------------------------------

typedef float v2f __attribute__((ext_vector_type(2)));

#define N_POINTS   (1u << 21)
#define NUM_LOD    16u
#define CB_SIZE    (1u << 19)          // 2^BAND_WIDTH
#define CB_MASK    (CB_SIZE - 1u)
#define PRIME0     265443567u
#define PRIME1     805459861u
#define BLOCK      256u
#define TOTAL_THREADS (N_POINTS * NUM_LOD)          // 2^25
#define PREFETCH_BLOCKS 2048u                       // 2048*256*128B = 64MB

__global__ __launch_bounds__(BLOCK) void hashgrid_encode_kernel(
    const v2f* __restrict__ pts,        // (2^21) float2
    const v2f* __restrict__ cb,         // (16 * 2^19) float2
    float* __restrict__ out,            // (2^21 * 32) float
    unsigned long long rw0, unsigned long long rw1,
    unsigned long long rw2, unsigned long long rw3)  // 16 x u16 resolutions
{
    const unsigned tid = blockIdx.x * BLOCK + threadIdx.x;

    // --- L2 warm: first 2048 blocks stream the 64MB codebook into L2 via
    // global_prefetch_b8 (one 128B line per lane, no counters, no VGPR return)
    if (blockIdx.x < PREFETCH_BLOCKS) {
        __builtin_prefetch((const char*)cb + (size_t)tid * 128u, /*rw=*/0, /*loc=*/3);
    }

    const unsigned n = tid >> 4;        // point index
    const unsigned l = tid & 15u;       // LOD index

    // Unpack this lane's resolution from the four packed u64 kernel args.
    const unsigned long long wsel =
        (l & 8u) ? ((l & 4u) ? rw3 : rw2)
                 : ((l & 4u) ? rw1 : rw0);
    const unsigned res = (unsigned)((wsel >> ((l & 3u) * 16u)) & 0xFFFFu);

    // Point coords (16 consecutive lanes share one point; HW merges the load).
    const v2f p = pts[n];
    const float s = (float)(res - 1u);
    const float x  = p.x * s;
    const float y  = p.y * s;
    const float xf = floorf(x);
    const float yf = floorf(y);
    const float fx = x - xf;
    const float fy = y - yf;
    const unsigned x0 = (unsigned)(int)xf;   // points in [0,1) -> x in [0,res-1)
    const unsigned y0 = (unsigned)(int)yf;

    unsigned id00, id10, id01, id11;
    if (res * res > CB_SIZE) {
        // Hashed level: int32 wrap-around mul/xor, floored mod 2^19 == & mask.
        const unsigned hx0 = x0 * PRIME0;
        const unsigned hx1 = hx0 + PRIME0;          // (x0+1)*PRIME0 mod 2^32
        const unsigned hy0 = y0 * PRIME1;
        const unsigned hy1 = hy0 + PRIME1;
        id00 = (hx0 ^ hy0) & CB_MASK;
        id10 = (hx1 ^ hy0) & CB_MASK;
        id01 = (hx0 ^ hy1) & CB_MASK;
        id11 = (hx1 ^ hy1) & CB_MASK;
    } else {
        // Dense level: row-major direct index.
        const unsigned base = x0 + y0 * res;
        id00 = base;
        id10 = base + 1u;
        id01 = base + res;
        id11 = base + res + 1u;
    }

    // Gather 4 corner float2 features (L2-resident after warm).
    const v2f* __restrict__ tbl = cb + (size_t)l * CB_SIZE;
    const v2f f00 = tbl[id00];
    const v2f f10 = tbl[id10];
    const v2f f01 = tbl[id01];
    const v2f f11 = tbl[id11];

    // Bilinear blend, same tap order as the reference einsum.
    const float gx = 1.0f - fx, gy = 1.0f - fy;
    const float w00 = gx * gy;
    const float w10 = fx * gy;
    const float w01 = gx * fy;
    const float w11 = fx * fy;

    v2f o;
    o.x = w00 * f00.x + w10 * f10.x + w01 * f01.x + w11 * f11.x;
    o.y = w00 * f00.y + w10 * f10.y + w01 * f01.y + w11 * f11.y;

    // out[n][2l .. 2l+1]  ==  out + 2*tid : coalesced, streaming -> NT store.
    __builtin_nontemporal_store(o, (v2f*)(out + 2u * (size_t)tid));
}

extern "C" void kernel_launch(void* const* d_in, const int* in_sizes, int n_in,
                              void* d_out, int out_size, void* d_ws, size_t ws_size,
                              hipStream_t stream) {
    (void)in_sizes; (void)n_in; (void)out_size; (void)d_ws; (void)ws_size;

    const v2f* pts = (const v2f*)d_in[0];   // points  : (2^21, 2) f32
    const v2f* cb  = (const v2f*)d_in[1];   // codebook: (16, 2^19, 2) f32
    float*     out = (float*)d_out;         // (2^21, 32) f32

    // Replicate numpy's fp64 scalar sequence exactly (same libm as the
    // harness): b = exp((log(2048)-log(16))/15); res_l = 1 + floor(16*b^l).
    // 16*b^15 is mathematically exactly 2048 -> the floor is ulp-sensitive,
    // so computing with identical libm calls is the only safe route.
    const double b = exp((log(2048.0) - log(16.0)) / 15.0);
    unsigned long long rw[4] = {0ull, 0ull, 0ull, 0ull};
    for (int l = 0; l < 16; ++l) {
        const int res = (int)(1.0 + floor(16.0 * pow(b, (double)l)));
        rw[l >> 2] |= (unsigned long long)((unsigned)res & 0xFFFFu)
                      << ((unsigned)(l & 3) * 16u);
    }

    const unsigned blocks = TOTAL_THREADS / BLOCK;   // 131072
    hashgrid_encode_kernel<<<dim3(blocks), dim3(BLOCK), 0, stream>>>(
        pts, cb, out, rw[0], rw[1], rw[2], rw[3]);
}